// simplicial_conv_71485435675277
// MI455X (gfx1250) — compile-verified
//
#include <hip/hip_runtime.h>
#include <hip/hip_bf16.h>
#include <math.h>

// Problem constants (from reference): x[8,4096,4096] f32, L[4096,4096] f32, W[16,4096] f32.
#define NN 4096
#define MB 8
#define KF 16
#define ROWS (MB * NN)          // 32768 flattened rows of X
#define ROWCHUNKS 64            // row split for colsum (64 rows per chunk)

typedef __attribute__((ext_vector_type(2))) float v2f;
typedef __attribute__((ext_vector_type(8))) float v8f;

// ---------------------------------------------------------------------------
// Kernel 1: partial column sums of L.  part[chunk][4096] = sum of 64 rows.
// Grid (4, 64) x 256 threads; thread handles 4 columns via float4 (coalesced).
// ---------------------------------------------------------------------------
__global__ __launch_bounds__(256) void colsum_partial(const float* __restrict__ L,
                                                      float* __restrict__ part) {
  const int cg = blockIdx.x * 256 + threadIdx.x;   // 0..1023 (group of 4 cols)
  const int chunk = blockIdx.y;                    // 0..63
  const float4* Lp = reinterpret_cast<const float4*>(L) +
                     (size_t)chunk * 64 * (NN / 4) + cg;
  float4 acc = make_float4(0.f, 0.f, 0.f, 0.f);
#pragma unroll 8
  for (int r = 0; r < 64; ++r) {
    float4 v = Lp[(size_t)r * (NN / 4)];
    acc.x += v.x; acc.y += v.y; acc.z += v.z; acc.w += v.w;
  }
  reinterpret_cast<float4*>(part)[(size_t)chunk * (NN / 4) + cg] = acc;
}

// ---------------------------------------------------------------------------
// Kernel 2: reduce partials -> col[4096].
// ---------------------------------------------------------------------------
__global__ __launch_bounds__(256) void colsum_reduce(const float* __restrict__ part,
                                                     float* __restrict__ col) {
  const int cg = blockIdx.x * 256 + threadIdx.x;   // 0..1023
  const float4* pp = reinterpret_cast<const float4*>(part) + cg;
  float4 acc = make_float4(0.f, 0.f, 0.f, 0.f);
#pragma unroll 8
  for (int c = 0; c < ROWCHUNKS; ++c) {
    float4 v = pp[(size_t)c * (NN / 4)];
    acc.x += v.x; acc.y += v.y; acc.z += v.z; acc.w += v.w;
  }
  reinterpret_cast<float4*>(col)[cg] = acc;
}

// ---------------------------------------------------------------------------
// Kernel 3: fused matvec (via f32 WMMA) + filter-tanh epilogue.
// One block = one 16-row tile of X[32768, 4096]; 4 waves split K=4096.
//
// WMMA mapping (V_WMMA_F32_16X16X4_F32, D = A(16x4) * B(4x16) + C):
//   A layout: v0 = {K0 | K2}, v1 = {K1 | K3} across low/high lane halves.
//   B layout: v0 = {K0 | K2}, v1 = {K1 | K3} rows striped across lanes.
// We permute K so a single float4 load per lane feeds two WMMAs:
//   low lanes carry K {k..k+3}, high lanes K {k+4..k+7}.
// B broadcasts col[k'] across all 16 columns, so every column of the
// accumulator holds the same s[i]; we read column 0 (lanes 0 and 16).
// ---------------------------------------------------------------------------
__global__ __launch_bounds__(128) void sconv_main(const float* __restrict__ x,
                                                  const float* __restrict__ col,
                                                  const float* __restrict__ W,
                                                  float* __restrict__ out) {
  __shared__ float sred[64];                 // [wave][16 rows]

  const int tile = blockIdx.x;               // 0..2047
  const int tid  = threadIdx.x;
  const int wave = tid >> 5;                 // 0..3 (wave32)
  const int lane = tid & 31;
  const int row  = (tile << 4) + (lane & 15);
  const int off  = (lane < 16) ? 0 : 4;      // low lanes K..K+3, high K+4..K+7
  const int kbeg = wave * (NN / 4);          // 1024 K per wave

  const float4* xp = reinterpret_cast<const float4*>(
      x + (size_t)row * NN + kbeg + off);
  const float4* cp = reinterpret_cast<const float4*>(col + kbeg + off);

  v8f c = {0.f, 0.f, 0.f, 0.f, 0.f, 0.f, 0.f, 0.f};

#pragma unroll 4
  for (int k = 0; k < NN / 4; k += 8) {      // 8 K-values per iteration
    float4 a = xp[0];  xp += 2;              // advance 8 floats
    float4 b = cp[0];  cp += 2;
    v2f a0 = {a.x, a.y};
    v2f a1 = {a.z, a.w};
    v2f b0 = {b.x, b.y};
    v2f b1 = {b.z, b.w};
    c = __builtin_amdgcn_wmma_f32_16x16x4_f32(false, a0, false, b0,
                                              (short)0, c, false, false);
    c = __builtin_amdgcn_wmma_f32_16x16x4_f32(false, a1, false, b1,
                                              (short)0, c, false, false);
  }

  // C/D 16x16 f32 layout: VGPR v, lanes 0-15 -> M=v (N=lane), lanes 16-31 -> M=v+8.
  // Column 0 lives in lane 0 (M=0..7) and lane 16 (M=8..15).
  if ((lane & 15) == 0) {
    const int base = wave * 16 + ((lane == 16) ? 8 : 0);
#pragma unroll
    for (int v = 0; v < 8; ++v) sred[base + v] = c[v];
  }
  __syncthreads();

  if (tid < 16) {
    const float s = sred[tid] + sred[16 + tid] + sred[32 + tid] + sred[48 + tid];
    const int r = (tile << 4) + tid;         // flattened row -> (b, i)
    const int i = r & (NN - 1);
    float acc = 0.f;
#pragma unroll
    for (int k = 0; k < KF; ++k) acc += tanhf(W[k * NN + i] * s);
    out[r] = acc;
  }
}

// ---------------------------------------------------------------------------
// Host launcher.  d_ws layout: col[4096] f32 at offset 0, partials[64][4096]
// f32 at offset 16 KB (total ~1.06 MB).
// ---------------------------------------------------------------------------
extern "C" void kernel_launch(void* const* d_in, const int* in_sizes, int n_in,
                              void* d_out, int out_size, void* d_ws, size_t ws_size,
                              hipStream_t stream) {
  const float* x = (const float*)d_in[0];   // [8, 4096, 4096]
  const float* L = (const float*)d_in[1];   // [4096, 4096]
  const float* W = (const float*)d_in[2];   // [16, 4096]
  float* out  = (float*)d_out;              // [8, 4096]
  float* col  = (float*)d_ws;               // 4096 f32
  float* part = (float*)d_ws + NN;          // 64*4096 f32

  colsum_partial<<<dim3(4, ROWCHUNKS), 256, 0, stream>>>(L, part);
  colsum_reduce<<<dim3(4), 256, 0, stream>>>(part, col);
  sconv_main<<<dim3(ROWS / 16), 128, 0, stream>>>(x, col, W, out);
}